// SetCriterion_52398601012070
// MI455X (gfx1250) — compile-verified
//
#include <hip/hip_runtime.h>
#include <hip/hip_bf16.h>
#include <math.h>

// ---------------------------------------------------------------------------
// SetCriterion losses on MI455X (gfx1250, wave32).
//   d_in[0] pred_logits      [32,256,21] f32
//   d_in[1] pred_polylines   [32,256,64,2] f32
//   d_in[2] target_polylines [32,128,64,2] f32
//   d_in[3] src_idx          [32,128] i32
//   d_in[4] labels           [32,128] i32
//   d_out   [3] f32 : {loss_ce, loss_polyline, loss_direction}
// Total traffic ~6.7MB, ~0.1 GFLOP -> latency bound; one wave32 per matched
// pair, polylines staged via CDNA5 async global->LDS loads (ASYNCcnt),
// deterministic two-phase reductions, final wave reduction on the WMMA pipe.
// ---------------------------------------------------------------------------

constexpr int B_ = 32, Q_ = 256, C1_ = 21, C_ = 20, M_ = 128, P_ = 64;
constexpr int ROWS  = B_ * Q_;   // 8192 CE rows
constexpr int PAIRS = B_ * M_;   // 4096 matched pairs

typedef __attribute__((ext_vector_type(2))) float v2f;
typedef __attribute__((ext_vector_type(8))) float v8f;
typedef int v4i_t __attribute__((vector_size(16)));  // matches builtin proto

// async builtin wants (v4i AS1*, v4i AS3*, imm, imm)
#define AS1V4(p) ((__attribute__((address_space(1))) v4i_t*)(p))
#define AS3V4(p) ((__attribute__((address_space(3))) v4i_t*)(p))

// ---------------- target-class construction (scatter, last write wins) -----
__global__ void k_init_classes(int* __restrict__ tc) {
  int i = blockIdx.x * blockDim.x + threadIdx.x;
  if (i < ROWS) tc[i] = C_;  // no-object class
}

__global__ void k_scatter_classes(int* __restrict__ tc,
                                  const int* __restrict__ src_idx,
                                  const int* __restrict__ labels) {
  int b = threadIdx.x;  // one lane per batch, serial over M -> in-order writes
  if (b < B_) {
    for (int m = 0; m < M_; ++m) {
      int q = src_idx[b * M_ + m];
      tc[b * Q_ + q] = labels[b * M_ + m];
    }
  }
}

// ---------------- cross entropy: one wave32 per row of 21 logits -----------
__global__ void k_ce(const float* __restrict__ logits,
                     const int* __restrict__ tc,
                     float* __restrict__ ws_ce) {
  const int lane = threadIdx.x & 31;
  const int wv   = threadIdx.x >> 5;
  const int row  = blockIdx.x * 8 + wv;  // grid sized exactly: no guard needed

  float x = (lane < C1_) ? logits[row * C1_ + lane] : -INFINITY;
  float mx = x;
  for (int s = 16; s; s >>= 1) mx = fmaxf(mx, __shfl_xor(mx, s, 32));
  float e = (lane < C1_) ? expf(x - mx) : 0.0f;
  float se = e;
  for (int s = 16; s; s >>= 1) se += __shfl_xor(se, s, 32);
  int t = tc[row];                 // wave-uniform
  float xt = __shfl(x, t, 32);     // logit of target class
  if (lane == 0) ws_ce[row] = -(xt - mx - logf(se));
}

// ---------------- chamfer + direction: one wave32 per (b,m) pair -----------
__global__ void k_chamfer(const float* __restrict__ pred_poly,
                          const float* __restrict__ tgt_poly,
                          const int* __restrict__ src_idx,
                          float* __restrict__ ws_pair,
                          float* __restrict__ ws_dir) {
  __shared__ __align__(16) float s_src[8][2 * P_];  // 512B per wave slice
  __shared__ __align__(16) float s_tgt[8][2 * P_];

  const int lane = threadIdx.x & 31;
  const int wv   = threadIdx.x >> 5;
  const int p    = blockIdx.x * 8 + wv;  // pair id, grid exact
  const int b    = p / M_;
  const int q    = src_idx[p];

  const float* gsrc = pred_poly + (size_t)(b * Q_ + q) * (2 * P_);
  const float* gtgt = tgt_poly  + (size_t)p * (2 * P_);

  // Stage both 512B polylines into this wave's LDS slice. Prefer the CDNA5
  // asynchronous global->LDS path (ASYNCcnt), fall back to register copy.
#if __has_builtin(__builtin_amdgcn_global_load_async_to_lds_b128)
  __builtin_amdgcn_global_load_async_to_lds_b128(
      AS1V4(gsrc + lane * 4), AS3V4(&s_src[wv][lane * 4]), 0, 0);
  __builtin_amdgcn_global_load_async_to_lds_b128(
      AS1V4(gtgt + lane * 4), AS3V4(&s_tgt[wv][lane * 4]), 0, 0);
  asm volatile("s_wait_asynccnt 0" ::: "memory");
#else
  float4 a = ((const float4*)gsrc)[lane];
  float4 c = ((const float4*)gtgt)[lane];
  ((float4*)&s_src[wv][0])[lane] = a;
  ((float4*)&s_tgt[wv][0])[lane] = c;
  // same-wave DS ops are in-order; no barrier needed (slice is wave-private)
#endif

  const float* S = s_src[wv];
  const float* T = s_tgt[wv];

  // lane owns target points {lane, lane+32} and source points {lane, lane+32}
  float tx0 = T[2 * lane],        ty0 = T[2 * lane + 1];
  float tx1 = T[2 * (lane + 32)], ty1 = T[2 * (lane + 32) + 1];
  float sx0 = S[2 * lane],        sy0 = S[2 * lane + 1];
  float sx1 = S[2 * (lane + 32)], sy1 = S[2 * (lane + 32) + 1];

  float m1a = INFINITY, m1b = INFINITY;  // min over src, per owned tgt point
  float m2a = INFINITY, m2b = INFINITY;  // min over tgt, per owned src point
  for (int i = 0; i < P_; ++i) {
    float sx = S[2 * i], sy = S[2 * i + 1];      // uniform LDS broadcast
    m1a = fminf(m1a, fabsf(sx - tx0) + fabsf(sy - ty0));
    m1b = fminf(m1b, fabsf(sx - tx1) + fabsf(sy - ty1));
    float tx = T[2 * i], ty = T[2 * i + 1];
    m2a = fminf(m2a, fabsf(sx0 - tx) + fabsf(sy0 - ty));
    m2b = fminf(m2b, fabsf(sx1 - tx) + fabsf(sy1 - ty));
  }
  float s1 = m1a + m1b;
  float s2 = m2a + m2b;
  for (int s = 16; s; s >>= 1) {
    s1 += __shfl_xor(s1, s, 32);
    s2 += __shfl_xor(s2, s, 32);
  }
  if (lane == 0) {
    ws_pair[p] = 0.5f * (s1 / (float)P_ + s2 / (float)P_);
    float sdx = S[2 * (P_ - 1)] - S[0], sdy = S[2 * (P_ - 1) + 1] - S[1];
    float tdx = T[2 * (P_ - 1)] - T[0], tdy = T[2 * (P_ - 1) + 1] - T[1];
    float ns = sqrtf(sdx * sdx + sdy * sdy) + 1e-6f;
    float nt = sqrtf(tdx * tdx + tdy * tdy) + 1e-6f;
    ws_dir[p] = 1.0f - (sdx * tdx + sdy * tdy) / (ns * nt);
  }
}

// ---------------- deterministic segmented reduction (WMMA-assisted) --------
// Wave reduction on the matrix pipe: D = A(16x4) x Ones(4x16) gives 16 row
// sums of A in one V_WMMA_F32_16X16X4_F32 (fixed hw accumulation order).
__global__ void k_reduce(const float* __restrict__ ws, float* __restrict__ out) {
  __shared__ float s_wave[8];
  const int seg = blockIdx.x;
  int off, n; float scale;
  if (seg == 0)      { off = 0;             n = ROWS;  scale = 1.0f / ROWS;  }
  else if (seg == 1) { off = ROWS;          n = PAIRS; scale = 1.0f / PAIRS; }
  else               { off = ROWS + PAIRS;  n = PAIRS; scale = 1.0f / PAIRS; }

  const float* v = ws + off;
  float acc = 0.0f;
  for (int i = threadIdx.x; i < n; i += 256) acc += v[i];  // fixed order

  // A: lane<16 holds A[m][0..1], lane>=16 holds A[m][2..3]; zeros in slot 1.
  v2f a;    a[0] = acc;  a[1] = 0.0f;
  v2f ones; ones[0] = 1.0f; ones[1] = 1.0f;
  v8f c = {};
  c = __builtin_amdgcn_wmma_f32_16x16x4_f32(
      false, a, false, ones, (short)0, c, false, false);
  // Each lane sums its 8 D rows (rows 0-7 on lanes 0-15, rows 8-15 on 16-31),
  // then one cross-half shuffle yields the 32-lane total in every lane.
  float half = c[0] + c[1] + c[2] + c[3] + c[4] + c[5] + c[6] + c[7];
  float tot  = half + __shfl_xor(half, 16, 32);

  const int lane = threadIdx.x & 31, wv = threadIdx.x >> 5;
  if (lane == 0) s_wave[wv] = tot;
  __syncthreads();
  if (threadIdx.x == 0) {
    float r = 0.0f;
    for (int w = 0; w < 8; ++w) r += s_wave[w];  // fixed order
    out[seg] = r * scale;
  }
}

// ---------------------------------------------------------------------------
extern "C" void kernel_launch(void* const* d_in, const int* in_sizes, int n_in,
                              void* d_out, int out_size, void* d_ws, size_t ws_size,
                              hipStream_t stream) {
  const float* pred_logits = (const float*)d_in[0];
  const float* pred_poly   = (const float*)d_in[1];
  const float* tgt_poly    = (const float*)d_in[2];
  const int*   src_idx     = (const int*)d_in[3];
  const int*   labels      = (const int*)d_in[4];
  float* out = (float*)d_out;

  // workspace layout: [ce partials | pair partials | dir partials | classes]
  float* ws_ce   = (float*)d_ws;
  float* ws_pair = ws_ce + ROWS;
  float* ws_dir  = ws_pair + PAIRS;
  int*   tc      = (int*)(ws_dir + PAIRS);

  k_init_classes   <<<(ROWS + 255) / 256, 256, 0, stream>>>(tc);
  k_scatter_classes<<<1, 32, 0, stream>>>(tc, src_idx, labels);
  k_ce             <<<ROWS / 8, 256, 0, stream>>>(pred_logits, tc, ws_ce);
  k_chamfer        <<<PAIRS / 8, 256, 0, stream>>>(pred_poly, tgt_poly, src_idx,
                                                   ws_pair, ws_dir);
  k_reduce         <<<3, 256, 0, stream>>>(ws_ce, out);
}